// MesoSep_68496138437437
// MI455X (gfx1250) — compile-verified
//
#include <hip/hip_runtime.h>

#define NN 50000
#define NE 800000
#define FD 128
#define DD 64
#define BN_EPS 1e-5f

typedef float v2f __attribute__((ext_vector_type(2)));
typedef float v4f __attribute__((ext_vector_type(4)));
typedef float v8f __attribute__((ext_vector_type(8)));

// ---------------------------------------------------------------------------
// WMMA 16xK * Kx64 micro-kernel (fp32, V_WMMA_F32_16X16X4_F32).
// A: 16 rows x K, row-major, stride K (LDS or global).
// BT: 64 cols x K (transposed weights), stride K (LDS).
// acc[c] covers output columns [c*16, c*16+16).
// fp32 A-frag layout: lanes 0-15 hold row=lane, {K=k,k+1}; lanes 16-31 {k+2,k+3}.
// fp32 B-frag layout: lanes 0-15 hold col=lane, {K=k,k+1}; lanes 16-31 {k+2,k+3}.
// C/D layout: VGPR r -> row r (lanes 0-15) / row r+8 (lanes 16-31), col = lane&15.
// ---------------------------------------------------------------------------
template <int K>
__device__ __forceinline__ void wmma_gemm16(const float* As, const float* BTs,
                                            const float* bias, v8f acc[4]) {
  const int lane = threadIdx.x & 31;
  const int half = lane >> 4;
  const int lid = lane & 15;
#pragma unroll
  for (int c = 0; c < 4; ++c) {
    float bv = bias[c * 16 + lid];
#pragma unroll
    for (int r = 0; r < 8; ++r) acc[c][r] = bv;
  }
#pragma unroll 4
  for (int k = 0; k < K; k += 4) {
    v2f a = *(const v2f*)(As + lid * K + k + 2 * half);
#pragma unroll
    for (int c = 0; c < 4; ++c) {
      v2f b = *(const v2f*)(BTs + (c * 16 + lid) * K + k + 2 * half);
      acc[c] = __builtin_amdgcn_wmma_f32_16x16x4_f32(
          false, a, false, b, (short)0, acc[c], false, false);
    }
  }
}

// Per-column sum / sum-of-squares into LDS (shared atomics).
__device__ __forceinline__ void acc_stats(const v8f acc[4], float* lsum, float* lsq) {
  const int lid = threadIdx.x & 15;
#pragma unroll
  for (int c = 0; c < 4; ++c) {
    float s = 0.f, q = 0.f;
#pragma unroll
    for (int r = 0; r < 8; ++r) {
      float v = acc[c][r];
      s += v;
      q += v * v;
    }
    atomicAdd(&lsum[c * 16 + lid], s);
    atomicAdd(&lsq[c * 16 + lid], q);
  }
}

// In-register BN+ReLU on C-layout accumulators (per-column scale/shift).
__device__ __forceinline__ void acc_bnrelu(v8f acc[4], const float* sc, const float* sh) {
  const int lid = threadIdx.x & 15;
#pragma unroll
  for (int c = 0; c < 4; ++c) {
    float scale = sc[c * 16 + lid];
    float shift = sh[c * 16 + lid];
#pragma unroll
    for (int r = 0; r < 8; ++r) acc[c][r] = fmaxf(acc[c][r] * scale + shift, 0.f);
  }
}

// Store C-layout accumulators into a 16x64 row-major tile (stride DD).
__device__ __forceinline__ void acc_store64(const v8f acc[4], float* row0) {
  const int lane = threadIdx.x & 31;
  const int half = lane >> 4;
  const int lid = lane & 15;
#pragma unroll
  for (int r = 0; r < 8; ++r) {
    float* p = row0 + (r + 8 * half) * DD;
#pragma unroll
    for (int c = 0; c < 4; ++c) p[c * 16 + lid] = acc[c][r];
  }
}

// ---------------------------------------------------------------------------
__global__ void k_zero(float* p, long n) {
  long i = (long)blockIdx.x * blockDim.x + threadIdx.x;
  long stride = (long)gridDim.x * blockDim.x;
  for (; i < n; i += stride) p[i] = 0.f;
}

__global__ void k_count(const int* __restrict__ dst, float* __restrict__ counts) {
  int e = blockIdx.x * blockDim.x + threadIdx.x;
  if (e < NE) atomicAdd(&counts[dst[e]], 1.0f);
}

// mean/var -> fused scale/shift:  y = x*scale + shift
__global__ void k_bn_finalize(const float* __restrict__ gstats_stage,
                              const float* __restrict__ g, const float* __restrict__ b,
                              float Minv, float* __restrict__ scsh_stage) {
  int t = threadIdx.x;
  if (t >= 2 * DD) return;
  int s = t >> 6, c = t & 63;
  const float* st = gstats_stage + s * 2 * DD;
  float mean = st[c] * Minv;
  float var = st[DD + c] * Minv - mean * mean;
  float rs = rsqrtf(var + BN_EPS);
  float scale = g[s * DD + c] * rs;
  float shift = b[s * DD + c] - mean * scale;
  float* o = scsh_stage + s * 2 * DD;
  o[c] = scale;
  o[DD + c] = shift;
}

// P1 = x @ fw1 + fb1 ; accumulate stage-0 stats.
__global__ __launch_bounds__(128) void k_node_gemm1(
    const float* __restrict__ x, const float* __restrict__ fw1,
    const float* __restrict__ fb1, float* __restrict__ P1, float* __restrict__ gstats) {
  __shared__ float Wt[DD * FD];
  __shared__ float lsum[DD], lsq[DD];
  const int s = blockIdx.y;
  const int tid = threadIdx.x;
  const float* W = fw1 + s * FD * DD;
  for (int i = tid; i < FD * DD; i += 128) {
    int k = i >> 6, n = i & 63;
    Wt[n * FD + k] = W[i];
  }
  if (tid < DD) { lsum[tid] = 0.f; lsq[tid] = 0.f; }
  __syncthreads();

  const int wave = tid >> 5;
  const int row0 = (blockIdx.x * 4 + wave) * 16;
  const bool active = row0 < NN;
  if (active) {
    v8f acc[4];
    wmma_gemm16<FD>(x + (long)row0 * FD, Wt, fb1 + s * DD, acc);
    acc_store64(acc, P1 + ((long)s * NN + row0) * DD);
    acc_stats(acc, lsum, lsq);
  }
  __syncthreads();
  float* gs = gstats + 0 * 256 + s * 128;
  if (tid < 64) atomicAdd(&gs[tid], lsum[tid]);
  else atomicAdd(&gs[tid], lsq[tid - 64]);
}

// P2 = relu(BN(P1)) @ fw2 + fb2 ; accumulate stage-1 stats.
__global__ __launch_bounds__(128) void k_node_gemm2(
    const float* __restrict__ P1, const float* __restrict__ fw2,
    const float* __restrict__ fb2, const float* __restrict__ scsh0,
    float* __restrict__ P2, float* __restrict__ gstats) {
  __shared__ float Wt[DD * DD];
  __shared__ float As[4][16 * DD];
  __shared__ float sc[DD], sh[DD];
  __shared__ float lsum[DD], lsq[DD];
  const int s = blockIdx.y;
  const int tid = threadIdx.x;
  const float* W = fw2 + s * DD * DD;
  for (int i = tid; i < DD * DD; i += 128) {
    int k = i >> 6, n = i & 63;
    Wt[n * DD + k] = W[i];
  }
  if (tid < DD) {
    sc[tid] = scsh0[s * 128 + tid];
    sh[tid] = scsh0[s * 128 + 64 + tid];
    lsum[tid] = 0.f;
    lsq[tid] = 0.f;
  }
  __syncthreads();

  const int wave = tid >> 5, lane = tid & 31;
  const int row0 = (blockIdx.x * 4 + wave) * 16;
  const bool active = row0 < NN;
  if (active) {
    const float* Pr = P1 + ((long)s * NN + row0) * DD;
    const int r = lane & 15, hh = lane >> 4;
#pragma unroll
    for (int j = 0; j < 8; ++j) {
      int cb = hh * 32 + j * 4;
      v4f p = *(const v4f*)(Pr + r * DD + cb);
      v4f o;
#pragma unroll
      for (int q = 0; q < 4; ++q) o[q] = fmaxf(p[q] * sc[cb + q] + sh[cb + q], 0.f);
      *(v4f*)(&As[wave][r * DD + cb]) = o;
    }
  }
  __syncthreads();
  if (active) {
    v8f acc[4];
    wmma_gemm16<DD>(&As[wave][0], Wt, fb2 + s * DD, acc);
    acc_store64(acc, P2 + ((long)s * NN + row0) * DD);
    acc_stats(acc, lsum, lsq);
  }
  __syncthreads();
  float* gs = gstats + 1 * 256 + s * 128;
  if (tid < 64) atomicAdd(&gs[tid], lsum[tid]);
  else atomicAdd(&gs[tid], lsq[tid - 64]);
}

// h = relu(BN(P2)) ; z = h @ lw0 + lb0
__global__ void k_node_finish(const float* __restrict__ P2, const float* __restrict__ scsh1,
                              const float* __restrict__ lw0, const float* __restrict__ lb0,
                              float* __restrict__ hbuf, float* __restrict__ zbuf) {
  const int s = blockIdx.y;
  const int n = blockIdx.x * blockDim.x + threadIdx.x;
  __shared__ float wz[DD], sc[DD], sh[DD];
  if (threadIdx.x < DD) {
    wz[threadIdx.x] = lw0[s * DD + threadIdx.x];
    sc[threadIdx.x] = scsh1[s * 128 + threadIdx.x];
    sh[threadIdx.x] = scsh1[s * 128 + 64 + threadIdx.x];
  }
  __syncthreads();
  if (n >= NN) return;
  const float* p = P2 + ((long)s * NN + n) * DD;
  float* hr = hbuf + ((long)s * NN + n) * DD;
  float acc = lb0[s];
  for (int c = 0; c < DD; c += 4) {
    v4f pv = *(const v4f*)(p + c);
    v4f hv;
#pragma unroll
    for (int q = 0; q < 4; ++q) {
      hv[q] = fmaxf(pv[q] * sc[c + q] + sh[c + q], 0.f);
      acc += hv[q] * wz[c + q];
    }
    *(v4f*)(hr + c) = hv;
  }
  zbuf[s * NN + n] = acc;
}

// Edge passes. MODE 1: gemm1 stats. MODE 2: gemm1->bn->gemm2 stats.
// MODE 3: full chain + atomic scatter into agg[dst].
template <int MODE>
__global__ __launch_bounds__(128) void k_edge(
    const float* __restrict__ hbuf, const int* __restrict__ ei,
    const float* __restrict__ ew1, const float* __restrict__ eb1,
    const float* __restrict__ ew2, const float* __restrict__ eb2,
    const float* __restrict__ scsh2, const float* __restrict__ scsh3,
    float* __restrict__ gstats, float* __restrict__ agg) {
  __shared__ float Wt1[DD * FD];                      // 32 KB
  __shared__ float As[4][16 * FD];                    // 32 KB
  __shared__ float Wt2[(MODE >= 2) ? DD * DD : 1];    // 16 KB
  __shared__ float As2[(MODE >= 2) ? 4 * 16 * DD : 1];// 16 KB
  __shared__ float sc2[DD], sh2[DD], sc3[DD], sh3[DD];
  __shared__ float lsum[DD], lsq[DD];

  const int s = blockIdx.y;
  const int tid = threadIdx.x;
  {
    const float* W = ew1 + s * FD * DD;
    for (int i = tid; i < FD * DD; i += 128) {
      int k = i >> 6, n = i & 63;
      Wt1[n * FD + k] = W[i];
    }
  }
  if constexpr (MODE >= 2) {
    const float* W = ew2 + s * DD * DD;
    for (int i = tid; i < DD * DD; i += 128) {
      int k = i >> 6, n = i & 63;
      Wt2[n * DD + k] = W[i];
    }
    if (tid < DD) {
      sc2[tid] = scsh2[s * 128 + tid];
      sh2[tid] = scsh2[s * 128 + 64 + tid];
    }
  }
  if constexpr (MODE == 3) {
    if (tid < DD) {
      sc3[tid] = scsh3[s * 128 + tid];
      sh3[tid] = scsh3[s * 128 + 64 + tid];
    }
  }
  if constexpr (MODE <= 2) {
    if (tid < DD) { lsum[tid] = 0.f; lsq[tid] = 0.f; }
  }
  __syncthreads();

  const int wave = tid >> 5, lane = tid & 31;
  const int e0 = (blockIdx.x * 4 + wave) * 16;
  const float* hb = hbuf + (long)s * NN * DD;

  {  // stage cat([x_i, x_j - x_i]) into LDS (L2-resident gathers of h)
    const int r = lane & 15, hh = lane >> 4;
    const int e = e0 + r;
    const int sn = ei[e];        // src  (x_j)
    const int dn = ei[NE + e];   // dst  (x_i)
    const float* xi = hb + (long)dn * DD;
    const float* xj = hb + (long)sn * DD;
    float* arow = &As[wave][r * FD];
#pragma unroll
    for (int j = 0; j < 8; ++j) {
      int cb = hh * 32 + j * 4;
      v4f a = *(const v4f*)(xi + cb);
      v4f b = *(const v4f*)(xj + cb);
      *(v4f*)(arow + cb) = a;
      v4f d;
#pragma unroll
      for (int q = 0; q < 4; ++q) d[q] = b[q] - a[q];
      *(v4f*)(arow + 64 + cb) = d;
    }
  }
  __syncthreads();

  v8f acc[4];
  wmma_gemm16<FD>(&As[wave][0], Wt1, eb1 + s * DD, acc);

  if constexpr (MODE == 1) {
    acc_stats(acc, lsum, lsq);
    __syncthreads();
    float* gs = gstats + 2 * 256 + s * 128;
    if (tid < 64) atomicAdd(&gs[tid], lsum[tid]);
    else atomicAdd(&gs[tid], lsq[tid - 64]);
  } else {
    acc_bnrelu(acc, sc2, sh2);
    acc_store64(acc, &As2[wave * 16 * DD]);
    __syncthreads();
    v8f acc2[4];
    wmma_gemm16<DD>(&As2[wave * 16 * DD], Wt2, eb2 + s * DD, acc2);
    if constexpr (MODE == 2) {
      acc_stats(acc2, lsum, lsq);
      __syncthreads();
      float* gs = gstats + 3 * 256 + s * 128;
      if (tid < 64) atomicAdd(&gs[tid], lsum[tid]);
      else atomicAdd(&gs[tid], lsq[tid - 64]);
    } else {
      acc_bnrelu(acc2, sc3, sh3);
      const int hh = lane >> 4, lid = lane & 15;
      float* ab = agg + (long)s * NN * DD;
#pragma unroll
      for (int r = 0; r < 8; ++r) {
        int e = e0 + r + 8 * hh;
        int d = ei[NE + e];
        float* arow = ab + (long)d * DD;
#pragma unroll
        for (int c = 0; c < 4; ++c) atomicAdd(&arow[c * 16 + lid], acc2[c][r]);
      }
    }
  }
}

__global__ void k_out(const float* __restrict__ zbuf, const float* __restrict__ agg,
                      const float* __restrict__ counts, const float* __restrict__ lw1,
                      const float* __restrict__ lb1, float* __restrict__ out) {
  const int s = blockIdx.y;
  const int n = blockIdx.x * blockDim.x + threadIdx.x;
  __shared__ float w[DD];
  if (threadIdx.x < DD) w[threadIdx.x] = lw1[s * DD + threadIdx.x];
  __syncthreads();
  if (n >= NN) return;
  const float* a = agg + ((long)s * NN + n) * DD;
  float acc = 0.f;
  for (int c = 0; c < DD; c += 4) {
    v4f av = *(const v4f*)(a + c);
#pragma unroll
    for (int q = 0; q < 4; ++q) acc += av[q] * w[c + q];
  }
  float cnt = fmaxf(counts[n], 1.0f);
  out[n * 2 + s] = zbuf[s * NN + n] + acc / cnt + lb1[s];
}

// ---------------------------------------------------------------------------
extern "C" void kernel_launch(void* const* d_in, const int* in_sizes, int n_in,
                              void* d_out, int out_size, void* d_ws, size_t ws_size,
                              hipStream_t stream) {
  const float* x = (const float*)d_in[0];
  const int* ei = (const int*)d_in[1];
  const float* fw1 = (const float*)d_in[2];
  const float* fb1 = (const float*)d_in[3];
  const float* fg1 = (const float*)d_in[4];
  const float* fbb1 = (const float*)d_in[5];
  const float* fw2 = (const float*)d_in[6];
  const float* fb2 = (const float*)d_in[7];
  const float* fg2 = (const float*)d_in[8];
  const float* fbb2 = (const float*)d_in[9];
  const float* lw0 = (const float*)d_in[10];
  const float* lb0 = (const float*)d_in[11];
  const float* ew1 = (const float*)d_in[12];
  const float* eb1 = (const float*)d_in[13];
  const float* eg1 = (const float*)d_in[14];
  const float* ebb1 = (const float*)d_in[15];
  const float* ew2 = (const float*)d_in[16];
  const float* eb2 = (const float*)d_in[17];
  const float* eg2 = (const float*)d_in[18];
  const float* ebb2 = (const float*)d_in[19];
  const float* lw1 = (const float*)d_in[20];
  const float* lb1 = (const float*)d_in[21];
  float* out = (float*)d_out;

  float* ws = (float*)d_ws;
  float* counts = ws;                                 // NN
  float* gstats = counts + NN;                        // 4 stages * 2 * 128 = 1024
  float* agg = gstats + 1024;                         // 2*NN*DD
  float* scsh = agg + (size_t)2 * NN * DD;            // 1024
  float* P1 = scsh + 1024;                            // 2*NN*DD
  float* P2 = P1 + (size_t)2 * NN * DD;               // 2*NN*DD
  float* hbuf = P2 + (size_t)2 * NN * DD;             // 2*NN*DD
  float* zbuf = hbuf + (size_t)2 * NN * DD;           // 2*NN

  const long zeroN = (long)NN + 1024 + (long)2 * NN * DD;
  k_zero<<<2048, 256, 0, stream>>>(ws, zeroN);
  k_count<<<(NE + 255) / 256, 256, 0, stream>>>(ei + NE, counts);

  dim3 gN((3125 + 3) / 4, 2);  // NN/16 row-tiles, 4 waves/block, 2 subnets
  k_node_gemm1<<<gN, 128, 0, stream>>>(x, fw1, fb1, P1, gstats);
  k_bn_finalize<<<1, 128, 0, stream>>>(gstats + 0, fg1, fbb1, 1.0f / NN, scsh + 0);
  k_node_gemm2<<<gN, 128, 0, stream>>>(P1, fw2, fb2, scsh + 0, P2, gstats);
  k_bn_finalize<<<1, 128, 0, stream>>>(gstats + 256, fg2, fbb2, 1.0f / NN, scsh + 256);

  dim3 gF((NN + 127) / 128, 2);
  k_node_finish<<<gF, 128, 0, stream>>>(P2, scsh + 256, lw0, lb0, hbuf, zbuf);

  dim3 gE(NE / 64, 2);  // 16-edge tiles, 4 waves/block
  k_edge<1><<<gE, 128, 0, stream>>>(hbuf, ei, ew1, eb1, ew2, eb2, scsh + 512, scsh + 768, gstats, agg);
  k_bn_finalize<<<1, 128, 0, stream>>>(gstats + 512, eg1, ebb1, 1.0f / NE, scsh + 512);
  k_edge<2><<<gE, 128, 0, stream>>>(hbuf, ei, ew1, eb1, ew2, eb2, scsh + 512, scsh + 768, gstats, agg);
  k_bn_finalize<<<1, 128, 0, stream>>>(gstats + 768, eg2, ebb2, 1.0f / NE, scsh + 768);
  k_edge<3><<<gE, 128, 0, stream>>>(hbuf, ei, ew1, eb1, ew2, eb2, scsh + 512, scsh + 768, gstats, agg);

  k_out<<<gF, 128, 0, stream>>>(zbuf, agg, counts, lw1, lb1, out);
}